// HEAT_36747740184624
// MI455X (gfx1250) — compile-verified
//
#include <hip/hip_runtime.h>
#include <math.h>

// ---------------------------------------------------------------------------
// HEATConv forward for MI455X (gfx1250), fp32 end-to-end.
// GEMMs: V_WMMA_F32_16X16X4_F32 (exact fp32), wave32, one 16-row tile/wave,
// persistent blocks, A-tiles staged global->LDS via async-LDS path when the
// toolchain exposes it (ASYNCcnt), LDS-resident weights.
// ---------------------------------------------------------------------------

typedef __attribute__((ext_vector_type(2))) float v2f;
typedef __attribute__((ext_vector_type(8))) float v8f;
typedef __attribute__((ext_vector_type(4))) int   v4i;

#define N1N 25000
#define N2N 25000
#define NN  50000
#define EE  200000
#define E2  400000
#define NGG 64

#if __has_builtin(__builtin_amdgcn_global_load_async_to_lds_b128)
#define HAVE_ASYNC_LDS 1
#else
#define HAVE_ASYNC_LDS 0
#endif

// 16-byte global -> LDS copy (direct async path when available)
__device__ __forceinline__ void copy16(const float* g, float* l)
{
#if HAVE_ASYNC_LDS
    __builtin_amdgcn_global_load_async_to_lds_b128(
        (__attribute__((address_space(1))) v4i*)g,
        (__attribute__((address_space(3))) v4i*)l, 0, 0);
#else
    *(float4*)l = *(const float4*)g;
#endif
}

__device__ __forceinline__ void stage_pre_wait()
{
#if HAVE_ASYNC_LDS
    // WAR: all prior DS reads of the tile must retire before async writes land
    asm volatile("s_wait_dscnt 0x0" ::: "memory");
#endif
}

__device__ __forceinline__ void stage_post_wait()
{
#if HAVE_ASYNC_LDS
    asm volatile("s_wait_asynccnt 0x0" ::: "memory");
#endif
}

// Stage a 16 x K1 fp32 tile (rows clamped to M-1) into LDS with row stride K1+4.
template<int K1>
__device__ __forceinline__ void stage_tile(const float* __restrict__ X, int M,
                                           int row0, float* A, int lane)
{
    constexpr int QP = K1 / 4;            // float4s per row
    stage_pre_wait();
    for (int it = 0; it < QP / 2; ++it) { // 16*QP quads / 32 lanes
        int i = lane + 32 * it;
        int r = i / QP, q = i % QP;
        int gr = row0 + r; if (gr >= M) gr = M - 1;
        copy16(X + (size_t)gr * K1 + q * 4, A + r * (K1 + 4) + q * 4);
    }
    stage_post_wait();
}

// ---- WMMA helper: C += A(16xK from LDS) * B(Kx16 slice of LDS weights) ----
__device__ __forceinline__ v8f wmma_acc(const float* A, int as,
                                        const float* W, int ws, int colBase,
                                        int kSteps, int lo, int hi, v8f c)
{
    for (int kb = 0; kb < kSteps; ++kb) {
        int k0 = kb * 4 + 2 * hi;                // A: lane<16 K=(0,1), lane>=16 K=(2,3)
        v2f a = *(const v2f*)(A + lo * as + k0); // row = lane&15
        v2f b;
        b.x = W[(k0    ) * ws + colBase + lo];   // B: VGPR0 = K rows {0|2}, VGPR1 = {1|3}
        b.y = W[(k0 + 1) * ws + colBase + lo];
        c = __builtin_amdgcn_wmma_f32_16x16x4_f32(false, a, false, b, (short)0, c,
                                                  false, false);
    }
    return c;
}

// ---- Generic fused 2-layer MLP: Y = act(X@W1 + b1) @ W2 + b2 (+ row dot) ----
template<int K1, int NOUT, bool LREL, bool DOT, bool BIAS>
__global__ __launch_bounds__(128)
void mlp2_kernel(const float* __restrict__ X, int M,
                 const float* __restrict__ W1g, const float* __restrict__ b1g,
                 const float* __restrict__ W2g, const float* __restrict__ b2g,
                 const float* __restrict__ w4g,
                 float* __restrict__ Y, float* __restrict__ dotOut)
{
    constexpr int AS = K1 + 4;   // 16B-aligned, even padded row stride
    constexpr int WS = 68;
    __shared__ __align__(16) float sW1[K1 * WS];
    __shared__ __align__(16) float sW2[64 * WS];
    __shared__ float sB1[64], sB2[64], sW4[64];
    __shared__ __align__(16) float sA[4][16 * AS];
    __shared__ __align__(16) float sT[4][16 * 68];

    for (int i = threadIdx.x; i < K1 * 64; i += 128)
        sW1[(i >> 6) * WS + (i & 63)] = W1g[i];
    for (int i = threadIdx.x; i < 64 * NOUT; i += 128)
        sW2[(i / NOUT) * WS + (i % NOUT)] = W2g[i];
    if (BIAS) {
        for (int i = threadIdx.x; i < 64; i += 128) {
            sB1[i] = b1g[i];
            sB2[i] = (i < NOUT) ? b2g[i] : 0.0f;
        }
    }
    if (DOT)
        for (int i = threadIdx.x; i < 64; i += 128) sW4[i] = w4g[i];
    __syncthreads();

    const int w = threadIdx.x >> 5, lane = threadIdx.x & 31;
    const int lo = lane & 15, hi = lane >> 4;
    const int tileCount = (M + 15) / 16;
    float* A = sA[w];
    float* T = sT[w];

    for (int tile = blockIdx.x * 4 + w; tile < tileCount; tile += gridDim.x * 4) {
        const int row0 = tile * 16;
        stage_tile<K1>(X, M, row0, A, lane);

        for (int ct = 0; ct < 4; ++ct) {
            v8f acc = {};
            acc = wmma_acc(A, AS, sW1, WS, ct * 16, K1 / 4, lo, hi, acc);
            for (int r = 0; r < 8; ++r) {
                float v = acc[r];
                if (BIAS) v += sB1[ct * 16 + lo];
                v = LREL ? (v > 0.0f ? v : 0.01f * v) : fmaxf(v, 0.0f);
                T[(hi * 8 + r) * 68 + ct * 16 + lo] = v;  // C layout: M = r + 8*hi
            }
        }

        if (DOT) {  // per-row dot of stage-1 output with w4 (attention "ea" term)
            float s = 0.0f;
            for (int c = hi * 32; c < hi * 32 + 32; ++c) s += T[lo * 68 + c] * sW4[c];
            s += __shfl_xor(s, 16, 32);
            if (hi == 0 && row0 + lo < M) dotOut[row0 + lo] = s;
        }

        for (int ct = 0; ct < NOUT / 16; ++ct) {
            v8f acc = {};
            acc = wmma_acc(T, 68, sW2, WS, ct * 16, 16, lo, hi, acc);
            for (int r = 0; r < 8; ++r) {
                int gr = row0 + hi * 8 + r;
                if (gr < M) {
                    float v = acc[r];
                    if (BIAS) v += sB2[ct * 16 + lo];
                    Y[(size_t)gr * NOUT + ct * 16 + lo] = v;
                }
            }
        }
    }
}

// ---- HeteroLinear: h = x @ W[node_type] + b[node_type] ----
__global__ __launch_bounds__(128)
void hetero_kernel(const float* __restrict__ X, const int* __restrict__ ntype, int M,
                   const float* __restrict__ Wg, const float* __restrict__ bg,
                   float* __restrict__ Hout)
{
    __shared__ __align__(16) float sW[2][64 * 68];
    __shared__ float sB[2][64];
    __shared__ __align__(16) float sA[4][16 * 68];
    for (int i = threadIdx.x; i < 2 * 64 * 64; i += 128)
        sW[i >> 12][((i >> 6) & 63) * 68 + (i & 63)] = Wg[i];
    for (int i = threadIdx.x; i < 128; i += 128) sB[i >> 6][i & 63] = bg[i];
    __syncthreads();

    const int w = threadIdx.x >> 5, lane = threadIdx.x & 31;
    const int lo = lane & 15, hi = lane >> 4;
    const int tileCount = (M + 15) / 16;
    float* A = sA[w];

    for (int tile = blockIdx.x * 4 + w; tile < tileCount; tile += gridDim.x * 4) {
        const int row0 = tile * 16;
        stage_tile<64>(X, M, row0, A, lane);
        for (int ct = 0; ct < 4; ++ct) {
            v8f c0 = {}; c0 = wmma_acc(A, 68, sW[0], 68, ct * 16, 16, lo, hi, c0);
            v8f c1 = {}; c1 = wmma_acc(A, 68, sW[1], 68, ct * 16, 16, lo, hi, c1);
            for (int r = 0; r < 8; ++r) {
                int gr = row0 + hi * 8 + r;
                if (gr < M) {
                    int t = ntype[gr];
                    float v = t ? (c1[r] + sB[1][ct * 16 + lo])
                                : (c0[r] + sB[0][ct * 16 + lo]);
                    Hout[(size_t)gr * 64 + ct * 16 + lo] = v;
                }
            }
        }
    }
}

// ---- Per-node prep: hmsg = h @ lin_w_top + lin_b ; hn1 = h.w_i ; hn2 = h.w_j ----
__global__ __launch_bounds__(128)
void nodeprep_kernel(const float* __restrict__ Hin, int M,
                     const float* __restrict__ Wg, const float* __restrict__ bg,
                     const float* __restrict__ v1g, const float* __restrict__ v2g,
                     float* __restrict__ hmsg, float* __restrict__ hn1,
                     float* __restrict__ hn2)
{
    __shared__ __align__(16) float sW[64 * 68];
    __shared__ float sB[64], sV1[64], sV2[64];
    __shared__ __align__(16) float sA[4][16 * 68];
    for (int i = threadIdx.x; i < 64 * 64; i += 128)
        sW[(i >> 6) * 68 + (i & 63)] = Wg[i];
    for (int i = threadIdx.x; i < 64; i += 128) {
        sB[i] = bg[i]; sV1[i] = v1g[i]; sV2[i] = v2g[i];
    }
    __syncthreads();

    const int w = threadIdx.x >> 5, lane = threadIdx.x & 31;
    const int lo = lane & 15, hi = lane >> 4;
    const int tileCount = (M + 15) / 16;
    float* A = sA[w];

    for (int tile = blockIdx.x * 4 + w; tile < tileCount; tile += gridDim.x * 4) {
        const int row0 = tile * 16;
        stage_tile<64>(Hin, M, row0, A, lane);

        float s1 = 0.0f, s2 = 0.0f;
        for (int c = hi * 32; c < hi * 32 + 32; ++c) {
            float a = A[lo * 68 + c];
            s1 += a * sV1[c]; s2 += a * sV2[c];
        }
        s1 += __shfl_xor(s1, 16, 32);
        s2 += __shfl_xor(s2, 16, 32);
        if (hi == 0 && row0 + lo < M) { hn1[row0 + lo] = s1; hn2[row0 + lo] = s2; }

        for (int ct = 0; ct < 4; ++ct) {
            v8f acc = {};
            acc = wmma_acc(A, 68, sW, 68, ct * 16, 16, lo, hi, acc);
            for (int r = 0; r < 8; ++r) {
                int gr = row0 + hi * 8 + r;
                if (gr < M)
                    hmsg[(size_t)gr * 64 + ct * 16 + lo] = acc[r] + sB[ct * 16 + lo];
            }
        }
    }
}

// ---- small helpers ----
__device__ __forceinline__ void atomicMaxF(float* addr, float val)
{
    if (val >= 0.0f) atomicMax((int*)addr, __float_as_int(val));
    else             atomicMin((unsigned int*)addr, (unsigned int)__float_as_int(val));
}

__device__ __forceinline__ void edgeSrcDst(int e, const int* ei0, const int* ei1,
                                           int& src, int& dst)
{
    if (e < EE) { src = ei0[e];      dst = ei0[EE + e]; }
    else        { src = ei1[e - EE]; dst = ei1[e];      }
}

__global__ void fill_kernel(float* p, float v, size_t n)
{
    size_t i = (size_t)blockIdx.x * blockDim.x + threadIdx.x;
    size_t st = (size_t)gridDim.x * blockDim.x;
    for (; i < n; i += st) p[i] = v;
}

__global__ void etdot_kernel(const float* __restrict__ et_emb,
                             const float* __restrict__ att_w,
                             float* __restrict__ etdot)
{
    __shared__ float s[128];
    int t = threadIdx.x;
    float e = et_emb[t];
    e = e > 0.0f ? e : 0.01f * e;                 // leaky_relu(et_emb, 0.01)
    s[t] = e * att_w[128 + (t & 63)];             // att_w rows [2H,3H)
    __syncthreads();
    for (int off = 32; off >= 1; off >>= 1) {
        if ((t & 63) < off) s[t] += s[t + off];
        __syncthreads();
    }
    if ((t & 63) == 0) etdot[t >> 6] = s[t];
}

__global__ void edge_logits_kernel(const int* __restrict__ ei0, const int* __restrict__ ei1,
                                   const int* __restrict__ etype,
                                   const float* __restrict__ hn1, const float* __restrict__ hn2,
                                   const float* __restrict__ attwea,
                                   const float* __restrict__ etdot,
                                   float* __restrict__ logits, float* __restrict__ segmax)
{
    int e = blockIdx.x * blockDim.x + threadIdx.x;
    if (e >= E2) return;
    int src, dst; edgeSrcDst(e, ei0, ei1, src, dst);
    float l = hn1[dst] + hn2[src] + etdot[etype[e]] + attwea[e];
    l = l > 0.0f ? l : 0.2f * l;                  // leaky_relu(.,0.2)
    logits[e] = l;
    atomicMaxF(&segmax[dst], l);
}

__global__ void edge_exp_kernel(const int* __restrict__ ei0, const int* __restrict__ ei1,
                                const float* __restrict__ logits,
                                const float* __restrict__ segmax,
                                float* __restrict__ eexp, float* __restrict__ segsum)
{
    int e = blockIdx.x * blockDim.x + threadIdx.x;
    if (e >= E2) return;
    int src, dst; edgeSrcDst(e, ei0, ei1, src, dst);
    (void)src;
    float ex = expf(logits[e] - segmax[dst]);
    eexp[e] = ex;
    atomicAdd(&segsum[dst], ex);
}

__global__ void edge_msg_kernel(const int* __restrict__ ei0, const int* __restrict__ ei1,
                                const float* __restrict__ eexp,
                                const float* __restrict__ segsum,
                                const float* __restrict__ hmsg,
                                const float* __restrict__ eamsg,
                                float* __restrict__ outacc)
{
    long long idx = (long long)blockIdx.x * blockDim.x + threadIdx.x;
    if (idx >= (long long)E2 * 64) return;
    int e = (int)(idx >> 6), c = (int)(idx & 63);
    int src, dst; edgeSrcDst(e, ei0, ei1, src, dst);
    float alpha = eexp[e] / (segsum[dst] + 1e-16f);
    float m = (hmsg[(size_t)src * 64 + c] + eamsg[(size_t)e * 64 + c]) * alpha;
    atomicAdd(&outacc[(size_t)dst * 64 + c], m);
}

__global__ void bn_stats_kernel(const float* __restrict__ xin, float* __restrict__ sums)
{
    __shared__ float sS[256], sQ[256];
    int c = threadIdx.x & 63, g = threadIdx.x >> 6;
    float s = 0.0f, q = 0.0f;
    for (int r = blockIdx.x * 4 + g; r < NN; r += gridDim.x * 4) {
        float v = xin[(size_t)r * 64 + c];
        s += v; q += v * v;
    }
    sS[threadIdx.x] = s; sQ[threadIdx.x] = q;
    __syncthreads();
    if (g == 0) {
        s = sS[c] + sS[64 + c] + sS[128 + c] + sS[192 + c];
        q = sQ[c] + sQ[64 + c] + sQ[128 + c] + sQ[192 + c];
        atomicAdd(&sums[c], s);
        atomicAdd(&sums[64 + c], q);
    }
}

__global__ void bn_apply_kernel(const float* __restrict__ xin, const float* __restrict__ sums,
                                const float* __restrict__ g, const float* __restrict__ b,
                                float* __restrict__ xout)
{
    int idx = blockIdx.x * blockDim.x + threadIdx.x;
    if (idx >= NN * 64) return;
    int c = idx & 63;
    float mu  = sums[c] * (1.0f / NN);
    float var = sums[64 + c] * (1.0f / NN) - mu * mu;   // biased, matches jnp.var
    xout[idx] = (xin[idx] - mu) * rsqrtf(var + 1e-5f) * g[c] + b[c];
}

__global__ void pool_kernel(const float* __restrict__ x, const int* __restrict__ batch,
                            float* __restrict__ emb)
{
    int idx = blockIdx.x * blockDim.x + threadIdx.x;
    if (idx >= NN * 64) return;
    int r = idx >> 6, c = idx & 63;
    atomicMaxF(&emb[batch[r] * 64 + c], x[idx]);
}

// ---------------------------------------------------------------------------
struct ConvP {
    const float *hl_w, *hl_b, *et_emb, *ea_w, *att_w, *lin_w, *lin_b, *bn_g, *bn_b;
};

extern "C" void kernel_launch(void* const* d_in, const int* in_sizes, int n_in,
                              void* d_out, int out_size, void* d_ws, size_t ws_size,
                              hipStream_t stream)
{
    (void)n_in; (void)out_size; (void)ws_size;

    const float *x_user, *x_item, *ea0, *ea1;
    const int *ei0, *ei1, *ntype, *etype, *batch;
    const float* mlpW[5][4];  // slots: node_user, node_item, edge0, edge1, out : W1,b1,W2,b2
    ConvP cp[2];

    bool insertion = (in_sizes[0] == N1N * 64);  // x_user first => insertion dict order
    if (insertion) {
        int i = 0;
        x_user = (const float*)d_in[i++]; x_item = (const float*)d_in[i++];
        ea0 = (const float*)d_in[i++];    ea1 = (const float*)d_in[i++];
        ei0 = (const int*)d_in[i++];      ei1 = (const int*)d_in[i++];
        ntype = (const int*)d_in[i++];    etype = (const int*)d_in[i++];
        batch = (const int*)d_in[i++];
        for (int m = 0; m < 5; ++m)
            for (int j = 0; j < 4; ++j) mlpW[m][j] = (const float*)d_in[i++];
        for (int l = 0; l < 2; ++l) {
            cp[l].hl_w  = (const float*)d_in[i++]; cp[l].hl_b  = (const float*)d_in[i++];
            cp[l].et_emb= (const float*)d_in[i++]; cp[l].ea_w  = (const float*)d_in[i++];
            cp[l].att_w = (const float*)d_in[i++]; cp[l].lin_w = (const float*)d_in[i++];
            cp[l].lin_b = (const float*)d_in[i++]; cp[l].bn_g  = (const float*)d_in[i++];
            cp[l].bn_b  = (const float*)d_in[i++];
        }
    } else {  // jax.tree_leaves sorted-key order
        batch = (const int*)d_in[0];
        ea0 = (const float*)d_in[1]; ea1 = (const float*)d_in[2];
        ei0 = (const int*)d_in[3];   ei1 = (const int*)d_in[4];
        etype = (const int*)d_in[5]; ntype = (const int*)d_in[6];
        int i = 7;
        for (int l = 0; l < 2; ++l) {
            cp[l].att_w = (const float*)d_in[i++]; cp[l].bn_b  = (const float*)d_in[i++];
            cp[l].bn_g  = (const float*)d_in[i++]; cp[l].ea_w  = (const float*)d_in[i++];
            cp[l].et_emb= (const float*)d_in[i++]; cp[l].hl_b  = (const float*)d_in[i++];
            cp[l].hl_w  = (const float*)d_in[i++]; cp[l].lin_b = (const float*)d_in[i++];
            cp[l].lin_w = (const float*)d_in[i++];
        }
        const int slot[5] = {2, 3, 1, 0, 4};  // edge0, edge1, node_item, node_user, out
        for (int k = 0; k < 5; ++k) {
            int m = slot[k];
            mlpW[m][0] = (const float*)d_in[i++];  // W1
            mlpW[m][2] = (const float*)d_in[i++];  // W2
            mlpW[m][1] = (const float*)d_in[i++];  // b1
            mlpW[m][3] = (const float*)d_in[i++];  // b2
        }
        x_item = (const float*)d_in[i++]; x_user = (const float*)d_in[i++];
    }

    // ---- workspace carve (≈258 MB) ----
    char* wptr = (char*)d_ws;
    auto alloc = [&](size_t nf) {
        float* p = (float*)wptr;
        wptr += ((nf * sizeof(float) + 255) / 256) * 256;
        return p;
    };
    float* x      = alloc((size_t)NN * 64);
    float* h      = alloc((size_t)NN * 64);
    float* hmsg   = alloc((size_t)NN * 64);
    float* outacc = alloc((size_t)NN * 64);
    float* edgeA  = alloc((size_t)E2 * 64);
    float* eamsg  = alloc((size_t)E2 * 64);
    float* hn1    = alloc(NN);
    float* hn2    = alloc(NN);
    float* attwea = alloc(E2);
    float* logits = alloc(E2);
    float* eexp   = alloc(E2);
    float* segmax = alloc(NN);
    float* segsum = alloc(NN);
    float* bnsums = alloc(128);
    float* etdot  = alloc(2);
    float* emb    = alloc((size_t)NGG * 64);

    dim3 blk(128);
    auto gemmGrid = [](int M) {
        unsigned b = (unsigned)(((M + 15) / 16 + 3) / 4);
        return dim3(b < 768u ? b : 768u);   // persistent blocks, grid-stride tiles
    };

    // ---- input MLPs ----
    mlp2_kernel<64, 64, false, false, true><<<gemmGrid(N1N), blk, 0, stream>>>(
        x_user, N1N, mlpW[0][0], mlpW[0][1], mlpW[0][2], mlpW[0][3], nullptr, x, nullptr);
    mlp2_kernel<32, 64, false, false, true><<<gemmGrid(N2N), blk, 0, stream>>>(
        x_item, N2N, mlpW[1][0], mlpW[1][1], mlpW[1][2], mlpW[1][3], nullptr,
        x + (size_t)N1N * 64, nullptr);
    mlp2_kernel<16, 64, false, false, true><<<gemmGrid(EE), blk, 0, stream>>>(
        ea0, EE, mlpW[2][0], mlpW[2][1], mlpW[2][2], mlpW[2][3], nullptr, edgeA, nullptr);
    mlp2_kernel<16, 64, false, false, true><<<gemmGrid(EE), blk, 0, stream>>>(
        ea1, EE, mlpW[3][0], mlpW[3][1], mlpW[3][2], mlpW[3][3], nullptr,
        edgeA + (size_t)EE * 64, nullptr);

    // ---- HEATConv layers ----
    for (int l = 0; l < 2; ++l) {
        const ConvP& p = cp[l];
        hetero_kernel<<<gemmGrid(NN), blk, 0, stream>>>(x, ntype, NN, p.hl_w, p.hl_b, h);
        etdot_kernel<<<1, 128, 0, stream>>>(p.et_emb, p.att_w, etdot);
        // fused: ea = lrelu(edge_attr@ea_w); eamsg = ea@lin_w[64:]; attwea = ea . att_w[192:]
        mlp2_kernel<64, 64, true, true, false><<<gemmGrid(E2), blk, 0, stream>>>(
            edgeA, E2, p.ea_w, nullptr, p.lin_w + 64 * 64, nullptr, p.att_w + 192,
            eamsg, attwea);
        nodeprep_kernel<<<gemmGrid(NN), blk, 0, stream>>>(
            h, NN, p.lin_w, p.lin_b, p.att_w, p.att_w + 64, hmsg, hn1, hn2);

        fill_kernel<<<256, 256, 0, stream>>>(segmax, -INFINITY, (size_t)NN);
        fill_kernel<<<256, 256, 0, stream>>>(segsum, 0.0f, (size_t)NN);
        fill_kernel<<<1024, 256, 0, stream>>>(outacc, 0.0f, (size_t)NN * 64);
        fill_kernel<<<1, 128, 0, stream>>>(bnsums, 0.0f, 128);

        edge_logits_kernel<<<(E2 + 255) / 256, 256, 0, stream>>>(
            ei0, ei1, etype, hn1, hn2, attwea, etdot, logits, segmax);
        edge_exp_kernel<<<(E2 + 255) / 256, 256, 0, stream>>>(
            ei0, ei1, logits, segmax, eexp, segsum);
        edge_msg_kernel<<<(unsigned)(((long long)E2 * 64 + 255) / 256), 256, 0, stream>>>(
            ei0, ei1, eexp, segsum, hmsg, eamsg, outacc);

        bn_stats_kernel<<<128, 256, 0, stream>>>(outacc, bnsums);
        bn_apply_kernel<<<(NN * 64 + 255) / 256, 256, 0, stream>>>(
            outacc, bnsums, p.bn_g, p.bn_b, x);
    }

    // ---- global max pool + output MLP ----
    fill_kernel<<<16, 256, 0, stream>>>(emb, -INFINITY, (size_t)NGG * 64);
    pool_kernel<<<(NN * 64 + 255) / 256, 256, 0, stream>>>(x, batch, emb);
    mlp2_kernel<64, 16, false, false, true><<<1, blk, 0, stream>>>(
        emb, NGG, mlpW[4][0], mlpW[4][1], mlpW[4][2], mlpW[4][3], nullptr,
        (float*)d_out, nullptr);
}